// MQJSLoss_80685255622904
// MI455X (gfx1250) — compile-verified
//
#include <hip/hip_runtime.h>

typedef __attribute__((ext_vector_type(2))) float v2f;
typedef __attribute__((ext_vector_type(8))) float v8f;

#define DIMD 256   // feature dim (compile-time; reference uses d=256)
#define BM   128   // block tile rows (i)
#define BN   64    // block tile cols (j)

// ---------------------------------------------------------------------------
// Kernel 1: per-row squared norms of x1, x2 and p=(x1+x2)/2 (p never stored).
// ---------------------------------------------------------------------------
__global__ __launch_bounds__(256) void mqjs_norms_kernel(
    const float* __restrict__ x1, const float* __restrict__ x2,
    float* __restrict__ s1, float* __restrict__ s2, float* __restrict__ sp)
{
    __shared__ float r0[256], r1[256], r2[256];
    const int row = blockIdx.x;
    const int t   = threadIdx.x;
    const size_t idx = (size_t)row * DIMD + t;
    const float a = x1[idx];
    const float b = x2[idx];
    const float p = 0.5f * (a + b);
    r0[t] = a * a; r1[t] = b * b; r2[t] = p * p;
    __syncthreads();
    for (int s = 128; s > 0; s >>= 1) {
        if (t < s) { r0[t] += r0[t + s]; r1[t] += r1[t + s]; r2[t] += r2[t + s]; }
        __syncthreads();
    }
    if (t == 0) { s1[row] = r0[0]; s2[row] = r1[0]; sp[row] = r2[0]; }
}

// ---------------------------------------------------------------------------
// Fragment bundle for one k-step (K=4): A rows from x1/x2, B rows from x1/x2.
// 32-bit offsets -> saddr + voffset addressing, k folds into the imm offset.
// ---------------------------------------------------------------------------
struct Frags {
    v2f a1[2], a2[2], b1[2], b2[2];
};

static __device__ __forceinline__ void load_frags(
    Frags& f, const float* __restrict__ x1, const float* __restrict__ x2,
    const unsigned oA[2], const unsigned oB[2], int k)
{
#pragma unroll
    for (int m = 0; m < 2; ++m) {
        f.a1[m] = *(const v2f*)(x1 + oA[m] + k);
        f.a2[m] = *(const v2f*)(x2 + oA[m] + k);
        f.b1[m] = *(const v2f*)(x1 + oB[m] + k);
        f.b2[m] = *(const v2f*)(x2 + oB[m] + k);
    }
}

static __device__ __forceinline__ void mma_step(
    const Frags& f, v8f accP[2][2], v8f accQ[2][2], v8f accS[2][2])
{
    v2f bp[2];
#pragma unroll
    for (int n = 0; n < 2; ++n)
        bp[n] = (f.b1[n] + f.b2[n]) * 0.5f;   // p fragment, in-register
#pragma unroll
    for (int m = 0; m < 2; ++m)
#pragma unroll
        for (int n = 0; n < 2; ++n) {
            accP[m][n] = __builtin_amdgcn_wmma_f32_16x16x4_f32(
                false, f.a1[m], false, bp[n], (short)0, accP[m][n], false, false);
            accQ[m][n] = __builtin_amdgcn_wmma_f32_16x16x4_f32(
                false, f.a2[m], false, bp[n], (short)0, accQ[m][n], false, false);
            accS[m][n] = __builtin_amdgcn_wmma_f32_16x16x4_f32(
                false, f.a1[m], false, f.b1[n], (short)0, accS[m][n], false, false);
        }
}

// ---------------------------------------------------------------------------
// Kernel 2: fused triple-GEMM + log epilogue, double-buffered k-loop.
// Block = 256 threads = 8 waves; wave grid 4(i) x 2(j); wave tile 32x32
// = 2x2 V_WMMA_F32_16X16X4_F32 tiles x 3 chains (x1.p, x2.p, x1.x1).
// ---------------------------------------------------------------------------
__global__ __launch_bounds__(256) void mqjs_main_kernel(
    const float* __restrict__ x1, const float* __restrict__ x2,
    const float* __restrict__ s1, const float* __restrict__ s2,
    const float* __restrict__ sp,
    float* __restrict__ partials, float wSe)
{
    const int tid  = threadIdx.x;
    const int wave = tid >> 5;
    const int lane = tid & 31;
    const int r    = lane & 15;   // row-within-fragment
    const int h    = lane >> 4;   // k-half selector

    const int i0 = blockIdx.x * BM + (wave & 3) * 32;
    const int j0 = blockIdx.y * BN + (wave >> 2) * 32;

    // 32-bit per-lane element offsets (max 8 MB buffers -> fits easily)
    unsigned oA[2], oB[2];
#pragma unroll
    for (int m = 0; m < 2; ++m) {
        oA[m] = (unsigned)(i0 + 16 * m + r) * DIMD + 2 * h;
        oB[m] = (unsigned)(j0 + 16 * m + r) * DIMD + 2 * h;
    }

    const v8f vzero = {0.f, 0.f, 0.f, 0.f, 0.f, 0.f, 0.f, 0.f};
    v8f accP[2][2], accQ[2][2], accS[2][2];
#pragma unroll
    for (int m = 0; m < 2; ++m)
#pragma unroll
        for (int n = 0; n < 2; ++n) { accP[m][n] = vzero; accQ[m][n] = vzero; accS[m][n] = vzero; }

    // Software-pipelined k-loop: loads for step k+4 issue before WMMAs on k.
    Frags f0, f1;
    load_frags(f0, x1, x2, oA, oB, 0);
    int k = 0;
    for (; k < DIMD - 8; k += 8) {
        load_frags(f1, x1, x2, oA, oB, k + 4);
        mma_step(f0, accP, accQ, accS);
        load_frags(f0, x1, x2, oA, oB, k + 8);
        mma_step(f1, accP, accQ, accS);
    }
    load_frags(f1, x1, x2, oA, oB, k + 4);
    mma_step(f0, accP, accQ, accS);
    mma_step(f1, accP, accQ, accS);

    // Epilogue: dist = si + sj - 2g ; contribution = log2(max(dist, 1))
    float sumCe = 0.f, sumSe = 0.f;
#pragma unroll
    for (int n = 0; n < 2; ++n) {
        const int j = j0 + 16 * n + r;
        const float spj = sp[j];
        const float s1j = s1[j];
#pragma unroll
        for (int m = 0; m < 2; ++m) {
#pragma unroll
            for (int v = 0; v < 8; ++v) {
                const int i = i0 + 16 * m + 8 * h + v;
                const float s1i = s1[i];
                const float s2i = s2[i];
                const float d1 = s1i + spj - 2.0f * accP[m][n][v];
                const float d2 = s2i + spj - 2.0f * accQ[m][n][v];
                const float d3 = s1i + s1j - 2.0f * accS[m][n][v];
                sumCe += __log2f(fmaxf(d1, 1.0f)) + __log2f(fmaxf(d2, 1.0f));
                sumSe += __log2f(fmaxf(d3, 1.0f));
            }
        }
    }
    const float local = sumCe - wSe * sumSe;

    // Deterministic in-block tree reduction -> one partial per block
    __shared__ float red[256];
    red[tid] = local;
    __syncthreads();
    for (int s = 128; s > 0; s >>= 1) {
        if (tid < s) red[tid] += red[tid + s];
        __syncthreads();
    }
    if (tid == 0) partials[blockIdx.y * gridDim.x + blockIdx.x] = red[0];
}

// ---------------------------------------------------------------------------
// Kernel 3: deterministic final reduction (double accumulation), scale,
// write scalar fp32 result.  scale = 0.5 * ln(2) / n
// ---------------------------------------------------------------------------
__global__ __launch_bounds__(256) void mqjs_reduce_kernel(
    const float* __restrict__ partials, int num, float* __restrict__ out, double scale)
{
    __shared__ double red[256];
    double s = 0.0;
    for (int i = threadIdx.x; i < num; i += 256) s += (double)partials[i];
    red[threadIdx.x] = s;
    __syncthreads();
    for (int st = 128; st > 0; st >>= 1) {
        if (threadIdx.x < st) red[threadIdx.x] += red[threadIdx.x + st];
        __syncthreads();
    }
    if (threadIdx.x == 0) out[0] = (float)(red[0] * scale);
}

extern "C" void kernel_launch(void* const* d_in, const int* in_sizes, int n_in,
                              void* d_out, int out_size, void* d_ws, size_t ws_size,
                              hipStream_t stream) {
    const float* x1 = (const float*)d_in[0];  // input  (n, d) fp32
    const float* x2 = (const float*)d_in[1];  // target (n, d) fp32
    const int N = in_sizes[0] / DIMD;         // n = 8192

    float* ws = (float*)d_ws;
    float* s1 = ws;                  // n floats
    float* s2 = ws + N;              // n floats
    float* sp = ws + 2 * (size_t)N;  // n floats
    float* partials = ws + 3 * (size_t)N;

    const int gi = N / BM;           // 64
    const int gj = N / BN;           // 128

    mqjs_norms_kernel<<<N, 256, 0, stream>>>(x1, x2, s1, s2, sp);

    dim3 grid(gi, gj);
    const float wSe = (float)N / (float)(N - 1);
    mqjs_main_kernel<<<grid, 256, 0, stream>>>(x1, x2, s1, s2, sp, partials, wSe);

    const double scale = 0.5 * 0.6931471805599453 / (double)N;
    mqjs_reduce_kernel<<<1, 256, 0, stream>>>(partials, gi * gj, (float*)d_out, scale);
}